// MultiLayerLoss_49031346651395
// MI455X (gfx1250) — compile-verified
//
#include <hip/hip_runtime.h>

typedef __attribute__((ext_vector_type(2))) float v2f;
typedef __attribute__((ext_vector_type(8))) float v8f;

// ---------- per-sample math ----------

__device__ __forceinline__ float ce2f(float a0, float a1, int y) {
    float m = fmaxf(a0, a1);
    float s = __expf(a0 - m) + __expf(a1 - m);
    float xy = (y == 0) ? a0 : a1;
    return __logf(s) + m - xy;
}
__device__ __forceinline__ float ce3f(float a0, float a1, float a2, int y) {
    float m = fmaxf(fmaxf(a0, a1), a2);
    float s = __expf(a0 - m) + __expf(a1 - m) + __expf(a2 - m);
    float xy = (y == 0) ? a0 : ((y == 1) ? a1 : a2);
    return __logf(s) + m - xy;
}
__device__ __forceinline__ float ce4f(float a0, float a1, float a2, float a3, int y) {
    float m = fmaxf(fmaxf(a0, a1), fmaxf(a2, a3));
    float s = __expf(a0 - m) + __expf(a1 - m) + __expf(a2 - m) + __expf(a3 - m);
    float xy = (y == 0) ? a0 : ((y == 1) ? a1 : ((y == 2) ? a2 : a3));
    return __logf(s) + m - xy;
}
// argmax, first occurrence wins on ties (matches jnp.argmax)
__device__ __forceinline__ int am2(float a0, float a1) { return (a1 > a0) ? 1 : 0; }
__device__ __forceinline__ int am3(float a0, float a1, float a2) {
    int i = 0; float b = a0;
    if (a1 > b) { i = 1; b = a1; }
    if (a2 > b) { i = 2; }
    return i;
}
__device__ __forceinline__ int am4(float a0, float a1, float a2, float a3) {
    int i = 0; float b = a0;
    if (a1 > b) { i = 1; b = a1; }
    if (a2 > b) { i = 2; b = a2; }
    if (a3 > b) { i = 3; }
    return i;
}
// hierarchy tables: layer1 {0:[0], 1:[1,2]}, layer2 {0:[0], 1:[1,3], 2:[2]}
__device__ __forceinline__ bool allowed1(int p, int c) { return (p == 0) ? (c == 0) : (c == 1 || c == 2); }
__device__ __forceinline__ bool allowed2(int p, int c) {
    return (p == 0) ? (c == 0) : ((p == 1) ? (c == 1 || c == 3) : (c == 2));
}
// 3.5^e - 1 for e in {0,1,2}
__device__ __forceinline__ float dterm(int e) { return (e == 0) ? 0.0f : ((e == 1) ? 2.5f : 11.25f); }

__device__ __forceinline__ void eval_sample(float p0a, float p0b,
                                            float p1a, float p1b, float p1c,
                                            float q0, float q1, float q2, float q3,
                                            int l0, int l1, int l2,
                                            float& ce_acc, float& d_acc) {
    ce_acc += ce2f(p0a, p0b, l0) + ce3f(p1a, p1b, p1c, l1) + ce4f(q0, q1, q2, q3, l2);
    int a0 = am2(p0a, p0b);
    int a1 = am3(p1a, p1b, p1c);
    int a2 = am4(q0, q1, q2, q3);
    int m0 = (a0 != l0) ? 1 : 0;
    int m1 = (a1 != l1) ? 1 : 0;
    int m2 = (a2 != l2) ? 1 : 0;
    int e1 = allowed1(a0, a1) ? 0 : (m0 + m1);
    int e2 = allowed2(a1, a2) ? 0 : (m1 + m2);
    d_acc += dterm(e1) + dterm(e2);
}

// ---------- kernels ----------

__global__ void mll_init(double* ws) {
    if (threadIdx.x < 2) ws[threadIdx.x] = 0.0;
}

__global__ __launch_bounds__(256) void mll_main(
    const float* __restrict__ pred0, const float* __restrict__ pred1,
    const float* __restrict__ pred2,
    const int* __restrict__ lab0, const int* __restrict__ lab1,
    const int* __restrict__ lab2,
    double* __restrict__ ws, int n) {
    const int tid = blockIdx.x * blockDim.x + threadIdx.x;
    const int stride = gridDim.x * blockDim.x;
    const int ng = n >> 2;  // groups of 4 samples -> all streams are float4/int4 aligned

    float ce_acc = 0.0f, d_acc = 0.0f;

    const float4* p0v = (const float4*)pred0;  // 2 x float4 per group
    const float4* p1v = (const float4*)pred1;  // 3 x float4 per group
    const float4* p2v = (const float4*)pred2;  // 4 x float4 per group
    const int4* l0v = (const int4*)lab0;
    const int4* l1v = (const int4*)lab1;
    const int4* l2v = (const int4*)lab2;

    for (int g = tid; g < ng; g += stride) {
        float4 a  = p0v[2 * g + 0];
        float4 b  = p0v[2 * g + 1];
        float4 c0 = p1v[3 * g + 0];
        float4 c1 = p1v[3 * g + 1];
        float4 c2 = p1v[3 * g + 2];
        float4 r0 = p2v[4 * g + 0];
        float4 r1 = p2v[4 * g + 1];
        float4 r2 = p2v[4 * g + 2];
        float4 r3 = p2v[4 * g + 3];
        int4 L0 = l0v[g];
        int4 L1 = l1v[g];
        int4 L2 = l2v[g];
        eval_sample(a.x, a.y, c0.x, c0.y, c0.z, r0.x, r0.y, r0.z, r0.w, L0.x, L1.x, L2.x, ce_acc, d_acc);
        eval_sample(a.z, a.w, c0.w, c1.x, c1.y, r1.x, r1.y, r1.z, r1.w, L0.y, L1.y, L2.y, ce_acc, d_acc);
        eval_sample(b.x, b.y, c1.z, c1.w, c2.x, r2.x, r2.y, r2.z, r2.w, L0.z, L1.z, L2.z, ce_acc, d_acc);
        eval_sample(b.z, b.w, c2.y, c2.z, c2.w, r3.x, r3.y, r3.z, r3.w, L0.w, L1.w, L2.w, ce_acc, d_acc);
    }
    // scalar tail (n % 4 != 0)
    for (int i = (ng << 2) + tid; i < n; i += stride) {
        eval_sample(pred0[2 * i], pred0[2 * i + 1],
                    pred1[3 * i], pred1[3 * i + 1], pred1[3 * i + 2],
                    pred2[4 * i], pred2[4 * i + 1], pred2[4 * i + 2], pred2[4 * i + 3],
                    lab0[i], lab1[i], lab2[i], ce_acc, d_acc);
    }

    // ---- block reduction: wave-level via V_WMMA_F32_16X16X4_F32, exact (0/1 multipliers) ----
    __shared__ float red[32];  // 8 waves x {c_half0, c_half1, d_half0, d_half1}
    const int lane = threadIdx.x & 31;
    const int wave = threadIdx.x >> 5;
    const int col = lane & 15;
    const int hv = lane >> 4;

#if __has_builtin(__builtin_amdgcn_wmma_f32_16x16x4_f32)
    // A (16x4): lane m<16: {K0,K1} = {ce,d}; lane m+16: {K2,K3} = {ce,d}
    // B (4x16): column 0 selects rows {0,2} (ce), column 1 selects rows {1,3} (d)
    // => D[m][0] = ce_m + ce_{m+16}, D[m][1] = d_m + d_{m+16}
    v2f av; av[0] = ce_acc; av[1] = d_acc;
    v2f bv; bv[0] = (col == 0) ? 1.0f : 0.0f; bv[1] = (col == 1) ? 1.0f : 0.0f;
    v8f acc = {};
    acc = __builtin_amdgcn_wmma_f32_16x16x4_f32(false, av, false, bv, (short)0, acc, false, false);
    // lane L holds column (L%16), rows (vgpr + 8*(L/16)); lanes {0,16} carry the two
    // halves of the ce column-sum, lanes {1,17} the d halves.
    float t = ((acc[0] + acc[1]) + (acc[2] + acc[3])) + ((acc[4] + acc[5]) + (acc[6] + acc[7]));
    if (col < 2) red[wave * 4 + col * 2 + hv] = t;
#else
    float c = ce_acc, d = d_acc;
    for (int off = 16; off > 0; off >>= 1) {
        c += __shfl_xor(c, off, 32);
        d += __shfl_xor(d, off, 32);
    }
    if (lane == 0) {
        red[wave * 4 + 0] = c; red[wave * 4 + 1] = 0.0f;
        red[wave * 4 + 2] = d; red[wave * 4 + 3] = 0.0f;
    }
#endif
    __syncthreads();
    if (threadIdx.x == 0) {
        double sc = 0.0, sd = 0.0;
        for (int w = 0; w < 8; ++w) {
            sc += (double)red[w * 4 + 0] + (double)red[w * 4 + 1];
            sd += (double)red[w * 4 + 2] + (double)red[w * 4 + 3];
        }
        __hip_atomic_fetch_add(&ws[0], sc, __ATOMIC_RELAXED, __HIP_MEMORY_SCOPE_AGENT);
        __hip_atomic_fetch_add(&ws[1], sd, __ATOMIC_RELAXED, __HIP_MEMORY_SCOPE_AGENT);
    }
}

__global__ void mll_fin(const double* __restrict__ ws, float* __restrict__ out, int n) {
    if (threadIdx.x == 0 && blockIdx.x == 0) {
        // lloss = ALPHA * (sum of three per-level CE sums) / B ; ALPHA = 1, BETA = 0.8
        double lloss = ws[0] / (double)n;
        out[0] = (float)(lloss + 0.8 * ws[1]);
    }
}

// ---------- launch ----------

extern "C" void kernel_launch(void* const* d_in, const int* in_sizes, int n_in,
                              void* d_out, int out_size, void* d_ws, size_t ws_size,
                              hipStream_t stream) {
    const float* pred0 = (const float*)d_in[0];
    const float* pred1 = (const float*)d_in[1];
    const float* pred2 = (const float*)d_in[2];
    const int* lab0 = (const int*)d_in[3];
    const int* lab1 = (const int*)d_in[4];
    const int* lab2 = (const int*)d_in[5];
    const int n = in_sizes[3];  // B

    double* ws = (double*)d_ws;
    float* out = (float*)d_out;

    int ng = n >> 2;
    int blocks = (ng + 255) / 256;
    if (blocks > 3072) blocks = 3072;
    if (blocks < 1) blocks = 1;

    hipLaunchKernelGGL(mll_init, dim3(1), dim3(32), 0, stream, ws);
    hipLaunchKernelGGL(mll_main, dim3(blocks), dim3(256), 0, stream,
                       pred0, pred1, pred2, lab0, lab1, lab2, ws, n);
    hipLaunchKernelGGL(mll_fin, dim3(1), dim3(32), 0, stream, ws, out, n);
}